// KineticOptimalDiscreteEulerSolver_29850022707391
// MI455X (gfx1250) — compile-verified
//
#include <hip/hip_runtime.h>
#include <hip/hip_bf16.h>

#define EPS 1e-8f

typedef float v2f __attribute__((ext_vector_type(2)));
typedef float v8f __attribute__((ext_vector_type(8)));

// One block = 16 tokens, 128 threads = 4 wave32s.
// Each wave handles 8 of the 32 column tiles (V=512 / 16).
// Per column tile: one V_WMMA_F32_16X16X4_F32 computes the rank-2 term
//   base[m, j] = a_m * u_j + b_m * v_j      (K=2, padded to 4 with zeros)
// then branchless VALU post-processing applies the one-hot corrections,
// clamp, and a selected per-tile reciprocal.
__global__ void __launch_bounds__(128)
kinetic_row_kernel(const float* __restrict__ sp,
                   const float* __restrict__ kp,
                   const float* __restrict__ dp,
                   const int*   __restrict__ xt,
                   const int*   __restrict__ x1,
                   float* __restrict__ out,
                   int V)
{
    const int tid   = threadIdx.x;
    const int wave  = tid >> 5;
    const int lane  = tid & 31;
    const int half  = lane >> 4;     // 0: lanes 0-15, 1: lanes 16-31
    const int l16   = lane & 15;
    const int tile0 = blockIdx.x * 16;   // first token of this block

    const float k   = kp[0];
    const float d   = dp[0];
    const float omk = 1.0f - k;

    __shared__ float a_s[16];    // p_t[i_n]
    __shared__ float b_s[16];    // p_dot[i_n]
    __shared__ float corr_s[16]; // a*d - b*k  (x1-column numerator fix)
    __shared__ int   x1_s[16];
    __shared__ int   xt_s[16];
    __shared__ float rs[16];     // per-token row sums

    if (tid < 16) {
        const int n   = tile0 + tid;
        const int iv  = xt[n];
        const int x1v = x1[n];
        const float spi = sp[iv];
        const float a = omk * spi + ((iv == x1v) ? k : 0.0f);
        const float b = d * (((iv == x1v) ? 1.0f : 0.0f) - spi);
        a_s[tid]    = a;
        b_s[tid]    = b;
        corr_s[tid] = a * d - b * k;
        x1_s[tid]   = x1v;
        xt_s[tid]   = iv;
        rs[tid]     = 0.0f;
    }
    __syncthreads();

    // A operand (constant across all column tiles):
    // lanes 0-15: K=0 -> a_M, K=1 -> b_M ; lanes 16-31: K=2,3 -> 0
    v2f A;
    A.x = (half == 0) ? a_s[l16] : 0.0f;
    A.y = (half == 0) ? b_s[l16] : 0.0f;

    // hoist per-token constants into registers (token = r + 8*half)
    float corr_r[8];
    int   x1_r[8], xt_r[8];
    float csum[8];
#pragma unroll
    for (int r = 0; r < 8; ++r) {
        const int tok = r + (half << 3);
        corr_r[r] = corr_s[tok];
        x1_r[r]   = x1_s[tok];
        xt_r[r]   = xt_s[tok];
        csum[r]   = 0.0f;
    }

    const int nTiles = V >> 4;          // 32
    for (int t = wave; t < nTiles; t += 4) {
        const int j   = (t << 4) + l16; // this lane's column
        const float spj = sp[j];
        const float cj  = omk * spj;    // base p_t[j] (no one-hot)

        // B operand: lanes 0-15: K=0 -> u_j = -d*sp[j], K=1 -> v_j = -c_j
        v2f B;
        B.x = (half == 0) ? (-d * spj) : 0.0f;
        B.y = (half == 0) ? (-cj)      : 0.0f;

        v8f C = {};
        C = __builtin_amdgcn_wmma_f32_16x16x4_f32(
                /*neg_a=*/false, A, /*neg_b=*/false, B,
                /*c_mod=*/(short)0, C, /*reuse_a=*/false, /*reuse_b=*/false);

        // only two possible denominators in this column: precompute both rcps
        const float inv0 = __builtin_amdgcn_rcpf(cj + EPS);      // j != x1
        const float invk = __builtin_amdgcn_rcpf(cj + k + EPS);  // j == x1

#pragma unroll
        for (int r = 0; r < 8; ++r) {
            const bool hx1 = (j == x1_r[r]);
            float base = C[r] + (hx1 ? corr_r[r] : 0.0f);
            float val  = fmaxf(base, 0.0f) * (hx1 ? invk : inv0);
            val = (j == xt_r[r]) ? 0.0f : val;   // diagonal forced to 0
            out[(size_t)(tile0 + r + (half << 3)) * V + j] = val;
            csum[r] += val;
        }
    }

    // one 16-lane xor-reduction per token at the end (not per tile)
#pragma unroll
    for (int r = 0; r < 8; ++r) {
        float s = csum[r];
        s += __shfl_xor(s, 1);
        s += __shfl_xor(s, 2);
        s += __shfl_xor(s, 4);
        s += __shfl_xor(s, 8);
        if (l16 == 0) atomicAdd(&rs[r + (half << 3)], s);
    }

    // ensure pass-1 stores (incl. the 0 at the diagonal) are complete,
    // then overwrite the diagonal with -rowsum.
    __threadfence_block();
    __syncthreads();

    if (tid < 16) {
        out[(size_t)(tile0 + tid) * V + xt_s[tid]] = -rs[tid];
    }
}

extern "C" void kernel_launch(void* const* d_in, const int* in_sizes, int n_in,
                              void* d_out, int out_size, void* d_ws, size_t ws_size,
                              hipStream_t stream) {
    (void)n_in; (void)out_size; (void)d_ws; (void)ws_size;
    const float* sp = (const float*)d_in[0];   // source_p [V]
    const float* kp = (const float*)d_in[1];   // k_t scalar
    const float* dp = (const float*)d_in[2];   // d_k_t scalar
    const int*   xt = (const int*)d_in[3];     // x_t [N]
    const int*   x1 = (const int*)d_in[4];     // x_1 [N]
    float* out = (float*)d_out;

    const int V = in_sizes[0];                 // 512
    const int N = in_sizes[3];                 // 512

    dim3 grid(N / 16);
    dim3 block(128);
    kinetic_row_kernel<<<grid, block, 0, stream>>>(sp, kp, dp, xt, x1, out, V);
}